// DeepFMModel_76570676953473
// MI455X (gfx1250) — compile-verified
//
#include <hip/hip_runtime.h>
#include <hip/hip_bf16.h>

typedef __attribute__((ext_vector_type(2))) float v2f;
typedef __attribute__((ext_vector_type(8))) float v8f;

#define BATCH 2048
#define NF 151193           // sum(VOCABS)+1
#define DEEP_K 80           // 77 padded to multiple of 4

// feature-index offsets into w_dense / fm_v
// VOCABS = [100000,50000,10,3,100,50,1000,24,5]
#define OFF_USER     0
#define OFF_ITEM     100000
#define OFF_AGE      150000
#define OFF_GENDER   150010
#define OFF_CITY     150013
#define OFF_CATEGORY 150113
#define OFF_BRAND    150163
#define OFF_HOUR     151163
#define OFF_DEVICE   151187
#define IDX_PRICE    (NF - 1)   // 151192

// ---------------------------------------------------------------------------
// Kernel 1: per-row gather. Computes logit_dense + logit_fm into partial[],
// and assembles the padded deep input row (77 real cols + 3 zero cols).
// One wave (32 lanes) per batch row.
// ---------------------------------------------------------------------------
__global__ void prep_kernel(
    const int* __restrict__ uid, const int* __restrict__ iid,
    const int* __restrict__ age, const int* __restrict__ gen,
    const int* __restrict__ city, const int* __restrict__ cat,
    const int* __restrict__ brand, const int* __restrict__ hour,
    const int* __restrict__ dev, const float* __restrict__ price,
    const float* __restrict__ e_user, const float* __restrict__ e_item,
    const float* __restrict__ e_age,  const float* __restrict__ e_gen,
    const float* __restrict__ e_city, const float* __restrict__ e_cat,
    const float* __restrict__ e_brand,const float* __restrict__ e_hour,
    const float* __restrict__ e_dev,
    const float* __restrict__ w_dense, const float* __restrict__ b_dense,
    const float* __restrict__ fm_v,
    float* __restrict__ deep_pad, float* __restrict__ partial)
{
    const int lane = threadIdx.x & 31;
    const int row  = blockIdx.x * (blockDim.x >> 5) + (threadIdx.x >> 5);
    if (row >= BATCH) return;

    const int id0 = uid[row], id1 = iid[row], id2 = age[row], id3 = gen[row];
    const int id4 = city[row], id5 = cat[row], id6 = brand[row], id7 = hour[row];
    const int id8 = dev[row];
    const float p  = price[row];
    const float pn = p * 0.001f;   // feat[:, NF-1] = price / 1000

    const int f0 = OFF_USER     + id0;
    const int f1 = OFF_ITEM     + id1;
    const int f2 = OFF_AGE      + id2;
    const int f3 = OFF_GENDER   + id3;
    const int f4 = OFF_CITY     + id4;
    const int f5 = OFF_CATEGORY + id5;
    const int f6 = OFF_BRAND    + id6;
    const int f7 = OFF_HOUR     + id7;
    const int f8 = OFF_DEVICE   + id8;

    // ---- FM 2nd order: lanes 0..15 each own one latent dim k ----
    float contrib = 0.f;
    if (lane < 16) {
        const int k = lane;
        float s = 0.f, sq = 0.f, v;
        v = fm_v[f0 * 16 + k]; s += v; sq += v * v;
        v = fm_v[f1 * 16 + k]; s += v; sq += v * v;
        v = fm_v[f2 * 16 + k]; s += v; sq += v * v;
        v = fm_v[f3 * 16 + k]; s += v; sq += v * v;
        v = fm_v[f4 * 16 + k]; s += v; sq += v * v;
        v = fm_v[f5 * 16 + k]; s += v; sq += v * v;
        v = fm_v[f6 * 16 + k]; s += v; sq += v * v;
        v = fm_v[f7 * 16 + k]; s += v; sq += v * v;
        v = fm_v[f8 * 16 + k]; s += v; sq += v * v;
        v = fm_v[IDX_PRICE * 16 + k]; s += pn * v; sq += pn * pn * v * v;
        contrib = 0.5f * (s * s - sq);
    }
    // reduce lanes 0..15 into lane 0 (lanes >=16 contribute 0)
    for (int o = 8; o >= 1; o >>= 1)
        contrib += __shfl_down(contrib, o, 32);

    if (lane == 0) {
        float ld = w_dense[f0] + w_dense[f1] + w_dense[f2] + w_dense[f3] +
                   w_dense[f4] + w_dense[f5] + w_dense[f6] + w_dense[f7] +
                   w_dense[f8] + pn * w_dense[IDX_PRICE] + b_dense[0];
        partial[row] = ld + contrib;
    }

    // ---- assemble padded deep input row (80 cols) ----
    // segment starts: user 0, item 16, age 32, gender 40, city 44,
    //                 category 52, brand 60, hour 68, device 72, price 76
    float* drow = deep_pad + (size_t)row * DEEP_K;
    for (int c = lane; c < DEEP_K; c += 32) {
        float val;
        if      (c < 16) val = e_user [id0 * 16 + c];
        else if (c < 32) val = e_item [id1 * 16 + (c - 16)];
        else if (c < 40) val = e_age  [id2 *  8 + (c - 32)];
        else if (c < 44) val = e_gen  [id3 *  4 + (c - 40)];
        else if (c < 52) val = e_city [id4 *  8 + (c - 44)];
        else if (c < 60) val = e_cat  [id5 *  8 + (c - 52)];
        else if (c < 68) val = e_brand[id6 *  8 + (c - 60)];
        else if (c < 72) val = e_hour [id7 *  4 + (c - 68)];
        else if (c < 76) val = e_dev  [id8 *  4 + (c - 72)];
        else if (c == 76) val = p;      // deep part uses raw price
        else val = 0.f;                 // pad cols 77..79
        drow[c] = val;
    }
}

// ---------------------------------------------------------------------------
// Kernel 2: zero-pad W1 (77x256) into W1p (80x256)
// ---------------------------------------------------------------------------
__global__ void pad_w1_kernel(const float* __restrict__ W1, float* __restrict__ W1p)
{
    int idx = blockIdx.x * blockDim.x + threadIdx.x;   // over 80*256
    if (idx >= DEEP_K * 256) return;
    int r = idx >> 8;                                  // /256
    W1p[idx] = (r < 77) ? W1[idx] : 0.f;
}

// ---------------------------------------------------------------------------
// Kernel 3: C[M,N] = relu(A[M,K] @ W[K,N] + bias[N]) using fp32 WMMA.
// N and K are compile-time template parameters: tile decomposition is
// shifts/masks, the K loop fully unrolls (K/4 wmmas per wave) so the
// compiler can clause the A/B loads and hide latency across many
// independent V_WMMA_F32_16X16X4_F32 issues.
// One wave per 16x16 output tile; grid sized so every wave owns a tile
// (EXEC stays all-ones, as WMMA requires).
// ---------------------------------------------------------------------------
template <int N, int K>
__global__ __launch_bounds__(256) void gemm_relu_wmma(
    const float* __restrict__ A,
    const float* __restrict__ W,
    const float* __restrict__ bias,
    float* __restrict__ C,
    int M)
{
    constexpr int NTILES = N >> 4;
    const int lane = threadIdx.x & 31;
    const int wid  = blockIdx.x * (blockDim.x >> 5) + (threadIdx.x >> 5);
    const int total = (M >> 4) * NTILES;
    if (wid >= total) return;

    const int mt = wid / NTILES;          // NTILES is a power of two -> shift
    const int nt = wid - mt * NTILES;
    const int sub  = lane & 15;
    const int half = lane >> 4;

    // A fragment: lane owns row m = mt*16+sub, K-pair {k0+2*half, k0+2*half+1}
    const float* Arow = A + (size_t)(mt * 16 + sub) * K + half * 2;
    // B fragment: lane owns col n = nt*16+sub, same K-pair, column-strided
    const float* Bcol = W + nt * 16 + sub;

    v8f acc = {};
#pragma unroll
    for (int k0 = 0; k0 < K; k0 += 4) {
        v2f a;
        a.x = Arow[k0];
        a.y = Arow[k0 + 1];
        const int kb = k0 + half * 2;
        v2f b;
        b.x = Bcol[(size_t)kb * N];
        b.y = Bcol[(size_t)(kb + 1) * N];
        // D = A*B + C, fp32 in / fp32 out
        acc = __builtin_amdgcn_wmma_f32_16x16x4_f32(
            /*neg_a=*/false, a, /*neg_b=*/false, b,
            /*c_mod=*/(short)0, acc, /*reuse_a=*/false, /*reuse_b=*/false);
    }

    const int n = nt * 16 + sub;
    const float bn = bias[n];
    const int mbase = mt * 16 + half * 8;
#pragma unroll
    for (int r = 0; r < 8; ++r) {
        float v = acc[r] + bn;
        v = fmaxf(v, 0.f);
        C[(size_t)(mbase + r) * N + n] = v;
    }
}

// ---------------------------------------------------------------------------
// Kernel 4: out[row] = partial[row] + H3[row,:] @ Wo + bo. One wave per row.
// ---------------------------------------------------------------------------
__global__ void out_kernel(const float* __restrict__ H3,
                           const float* __restrict__ Wo,
                           const float* __restrict__ bo,
                           const float* __restrict__ partial,
                           float* __restrict__ out)
{
    const int lane = threadIdx.x & 31;
    const int row  = blockIdx.x * (blockDim.x >> 5) + (threadIdx.x >> 5);
    if (row >= BATCH) return;

    const float* h = H3 + (size_t)row * 64;
    float s = h[lane] * Wo[lane] + h[lane + 32] * Wo[lane + 32];
    for (int o = 16; o >= 1; o >>= 1)
        s += __shfl_down(s, o, 32);
    if (lane == 0)
        out[row] = partial[row] + s + bo[0];
}

// ---------------------------------------------------------------------------
extern "C" void kernel_launch(void* const* d_in, const int* in_sizes, int n_in,
                              void* d_out, int out_size, void* d_ws, size_t ws_size,
                              hipStream_t stream)
{
    (void)in_sizes; (void)n_in; (void)out_size; (void)ws_size;

    const int*   uid   = (const int*)d_in[0];
    const int*   iid   = (const int*)d_in[1];
    const int*   age   = (const int*)d_in[2];
    const int*   gen   = (const int*)d_in[3];
    const int*   city  = (const int*)d_in[4];
    const int*   cat   = (const int*)d_in[5];
    const int*   brand = (const int*)d_in[6];
    const int*   hour  = (const int*)d_in[7];
    const int*   dev   = (const int*)d_in[8];
    const float* price = (const float*)d_in[9];
    const float* e_user  = (const float*)d_in[10];
    const float* e_item  = (const float*)d_in[11];
    const float* e_age   = (const float*)d_in[12];
    const float* e_gen   = (const float*)d_in[13];
    const float* e_city  = (const float*)d_in[14];
    const float* e_cat   = (const float*)d_in[15];
    const float* e_brand = (const float*)d_in[16];
    const float* e_hour  = (const float*)d_in[17];
    const float* e_dev   = (const float*)d_in[18];
    const float* w_dense = (const float*)d_in[19];
    const float* b_dense = (const float*)d_in[20];
    const float* fm_v    = (const float*)d_in[21];
    const float* W1 = (const float*)d_in[22];
    const float* b1 = (const float*)d_in[23];
    const float* W2 = (const float*)d_in[24];
    const float* b2 = (const float*)d_in[25];
    const float* W3 = (const float*)d_in[26];
    const float* b3 = (const float*)d_in[27];
    const float* Wo = (const float*)d_in[28];
    const float* bo = (const float*)d_in[29];

    float* out = (float*)d_out;

    // workspace carve-up (all fp32)
    float* ws       = (float*)d_ws;
    float* deep_pad = ws;                                // 2048*80
    float* W1p      = deep_pad + (size_t)BATCH * DEEP_K; // 80*256
    float* H1       = W1p + DEEP_K * 256;                // 2048*256
    float* H2       = H1 + (size_t)BATCH * 256;          // 2048*128
    float* H3       = H2 + (size_t)BATCH * 128;          // 2048*64
    float* partial  = H3 + (size_t)BATCH * 64;           // 2048

    // 1) gather + FM + assemble deep input (8 rows per 256-thread block)
    prep_kernel<<<BATCH / 8, 256, 0, stream>>>(
        uid, iid, age, gen, city, cat, brand, hour, dev, price,
        e_user, e_item, e_age, e_gen, e_city, e_cat, e_brand, e_hour, e_dev,
        w_dense, b_dense, fm_v, deep_pad, partial);

    // 2) pad W1 to 80 rows
    pad_w1_kernel<<<(DEEP_K * 256 + 255) / 256, 256, 0, stream>>>(W1, W1p);

    // 3) MLP layers via fp32 WMMA (8 tiles per 256-thread block)
    // L1: [2048x80] @ [80x256]  -> 128*16 = 2048 tiles
    gemm_relu_wmma<256, DEEP_K><<<2048 / 8, 256, 0, stream>>>(
        deep_pad, W1p, b1, H1, BATCH);
    // L2: [2048x256] @ [256x128] -> 128*8 = 1024 tiles
    gemm_relu_wmma<128, 256><<<1024 / 8, 256, 0, stream>>>(
        H1, W2, b2, H2, BATCH);
    // L3: [2048x128] @ [128x64]  -> 128*4 = 512 tiles
    gemm_relu_wmma<64, 128><<<512 / 8, 256, 0, stream>>>(
        H2, W3, b3, H3, BATCH);

    // 4) final projection + combine logits
    out_kernel<<<BATCH / 8, 256, 0, stream>>>(H3, Wo, bo, partial, out);
}